// LCC_25898652795407
// MI455X (gfx1250) — compile-verified
//
#include <hip/hip_runtime.h>

// LCC loss for (2,1,160,160,160) f32 volumes on MI455X (gfx1250, wave32).
// Separable 5-tap box filters. Y/Z passes use V_WMMA_F32_16X16X4_F32:
//   Out(16 outputs x 16 x-lanes) = Band(16x20 ones-band) x In(20 lines x 16 x)
// accumulated over K=20 as five 16x16x4 f32 WMMAs. X passes are float4 VALU.
// Y pass stages its input tile into LDS with global_load_async_to_lds_b64
// (ASYNCcnt) when the toolchain exposes the builtin.

#define DIM   160
#define HWu   25600u
#define DHWu  4096000u
#define TOTALu 8192000u                 // 2 * 160^3 elements
#define SZ_INV (1.0f / 125.0f)
#define EPSV   1e-10f
#define X4BLOCKS 8000                   // (TOTAL/4) / 256
#define WBLOCKS  4000                   // 32000 tiles / 8 waves per block

typedef __attribute__((ext_vector_type(2))) float v2f;
typedef __attribute__((ext_vector_type(8))) float v8f;
typedef __attribute__((ext_vector_type(2))) int   i2;

#ifndef __has_builtin
#define __has_builtin(x) 0
#endif
#if __has_builtin(__builtin_amdgcn_global_load_async_to_lds_b64)
#define USE_ASYNC_LDS 1
typedef i2 __attribute__((address_space(1)))* g64p;   // global <2 x i32>* (b64 payload)
typedef i2 __attribute__((address_space(3)))* l64p;   // LDS    <2 x i32>*
__device__ __forceinline__ void wait_async0() {
#if __has_builtin(__builtin_amdgcn_s_wait_asynccnt)
  __builtin_amdgcn_s_wait_asynccnt(0);
#else
  asm volatile("s_wait_asynccnt 0x0" ::: "memory");
#endif
}
#else
#define USE_ASYNC_LDS 0
#endif

// ---- core: 16-output 5-tap box sum along a strided dim, via 5x WMMA f32 16x16x4.
// A (16x4 chunk): lane = row m, VGPR v holds K = v + 2*(lane>=16); band = 31u<<m.
// B (4x16 chunk): lane&15 = col n (contiguous x), VGPR v holds K = v + 2*(lane>=16).
// D (16x16): VGPR r -> output line m = r + 8*(lane>=16), col n = lane&15.
__device__ __forceinline__ v8f wave_box16_g(const float* __restrict__ in,
                                            unsigned base, int f0, unsigned fstride) {
  const int lane = __builtin_amdgcn_workitem_id_x() & 31;
  const int mn   = lane & 15;
  const int h    = lane >> 4;
  const unsigned band = 31u << mn;      // ones band: out[m] = sum in[m..m+4]
  v8f acc = {};
#pragma unroll
  for (int c = 0; c < 5; ++c) {
    v2f a, b;
#pragma unroll
    for (int v = 0; v < 2; ++v) {
      const int kg = 4 * c + v + 2 * h;               // global K in 0..19
      a[v] = (float)((band >> kg) & 1u);
      int j = f0 - 2 + kg;                            // replicate pad via clamp
      j = j < 0 ? 0 : (j > DIM - 1 ? DIM - 1 : j);
      b[v] = in[base + (unsigned)j * fstride + (unsigned)mn];
    }
    acc = __builtin_amdgcn_wmma_f32_16x16x4_f32(false, a, false, b,
                                                (short)0, acc, false, false);
  }
  return acc;
}

// tile id -> base offset (incl. x0) and f0. 10 x-tiles, 10 f-tiles, 320 outer lines.
__device__ __forceinline__ unsigned tile_decode(int t, unsigned mulLo, int& f0) {
  const int xt = t % 10;
  const int ft = (t / 10) % 10;
  const int o  = t / 100;               // 0..319 = batch*160 + other-dim index
  f0 = ft * 16;
  return (unsigned)(o / DIM) * DHWu + (unsigned)(o % DIM) * mulLo + (unsigned)(xt * 16);
}

// ---- X-direction box sum: 4 outputs/thread, b128 loads/stores, replicate edges
__global__ void __launch_bounds__(256) kbox_x(const float* __restrict__ in,
                                              float* __restrict__ out) {
  const unsigned q  = blockIdx.x * 256u + threadIdx.x;    // quad id
  const unsigned row = q / 40u;
  const unsigned x0  = (q - row * 40u) * 4u;
  const unsigned rb  = row * 160u;
  const float4 a = *(const float4*)(in + rb + x0);
  const unsigned lb = (x0 == 0u)   ? 0u   : x0 - 4u;
  const unsigned rx = (x0 == 156u) ? 156u : x0 + 4u;
  const float4 L = *(const float4*)(in + rb + lb);
  const float4 R = *(const float4*)(in + rb + rx);
  const float l2 = (x0 == 0u)   ? a.x : L.z;
  const float l1 = (x0 == 0u)   ? a.x : L.w;
  const float r1 = (x0 == 156u) ? a.w : R.x;
  const float r2 = (x0 == 156u) ? a.w : R.y;
  float4 o;
  o.x = l2 + l1 + a.x + a.y + a.z;
  o.y = l1 + a.x + a.y + a.z + a.w;
  o.z = a.x + a.y + a.z + a.w + r1;
  o.w = a.y + a.z + a.w + r1 + r2;
  *(float4*)(out + rb + x0) = o;
}

// X box sum of squared input (fuses t = c^2 into the first chain-2 pass)
__global__ void __launch_bounds__(256) kbox_x_sq(const float* __restrict__ in,
                                                 float* __restrict__ out) {
  const unsigned q  = blockIdx.x * 256u + threadIdx.x;
  const unsigned row = q / 40u;
  const unsigned x0  = (q - row * 40u) * 4u;
  const unsigned rb  = row * 160u;
  const float4 a = *(const float4*)(in + rb + x0);
  const unsigned lb = (x0 == 0u)   ? 0u   : x0 - 4u;
  const unsigned rx = (x0 == 156u) ? 156u : x0 + 4u;
  const float4 L = *(const float4*)(in + rb + lb);
  const float4 R = *(const float4*)(in + rb + rx);
  const float l2 = (x0 == 0u)   ? a.x : L.z;
  const float l1 = (x0 == 0u)   ? a.x : L.w;
  const float r1 = (x0 == 156u) ? a.w : R.x;
  const float r2 = (x0 == 156u) ? a.w : R.y;
  const float ax = a.x * a.x, ay = a.y * a.y, az = a.z * a.z, aw = a.w * a.w;
  const float s2 = l2 * l2, s1 = l1 * l1, t1 = r1 * r1, t2 = r2 * r2;
  float4 o;
  o.x = s2 + s1 + ax + ay + az;
  o.y = s1 + ax + ay + az + aw;
  o.z = ax + ay + az + aw + t1;
  o.w = ay + az + aw + t1 + t2;
  *(float4*)(out + rb + x0) = o;
}

// ---- Y-pass WMMA box (fstride=160, mulLo=HW); async LDS tile staging when available
__global__ void __launch_bounds__(256) kbox_wmma(const float* __restrict__ in,
                                                 float* __restrict__ out,
                                                 unsigned fstride, unsigned mulLo) {
  const int wave = threadIdx.x >> 5;
  const int t = blockIdx.x * 8 + wave;
  int f0; const unsigned base = tile_decode(t, mulLo, f0);
  const int lane = threadIdx.x & 31;
  const int mn = lane & 15, h = lane >> 4;
  v8f acc = {};
#if USE_ASYNC_LDS
  // Stage the 20-line x 16-float tile into LDS: 5 async b64 fills per wave
  // (lane -> line group l>>3, 8-byte pair l&7), then s_wait_asynccnt 0.
  __shared__ float tile[8 * 20 * 16];
  float* wt = tile + wave * 320;
  {
    const int sub  = lane & 7;
    const int kgrp = lane >> 3;
#pragma unroll
    for (int i = 0; i < 5; ++i) {
      const int kg = i * 4 + kgrp;
      int j = f0 - 2 + kg;
      j = j < 0 ? 0 : (j > DIM - 1 ? DIM - 1 : j);
      const float* g = in + base + (unsigned)j * fstride + (unsigned)(sub * 2);
      __builtin_amdgcn_global_load_async_to_lds_b64(
          (g64p)g, (l64p)(wt + kg * 16 + sub * 2), 0, 0);
    }
    wait_async0();
  }
  const unsigned band = 31u << mn;
#pragma unroll
  for (int c = 0; c < 5; ++c) {
    v2f a, b;
#pragma unroll
    for (int v = 0; v < 2; ++v) {
      const int kg = 4 * c + v + 2 * h;
      a[v] = (float)((band >> kg) & 1u);
      b[v] = wt[kg * 16 + mn];
    }
    acc = __builtin_amdgcn_wmma_f32_16x16x4_f32(false, a, false, b,
                                                (short)0, acc, false, false);
  }
#else
  acc = wave_box16_g(in, base, f0, fstride);
#endif
#pragma unroll
  for (int r = 0; r < 8; ++r)
    out[base + (unsigned)(f0 + r + 8 * h) * fstride + (unsigned)mn] = acc[r];
}

// ---- Z pass + finish local mean: c = orig - box/125
__global__ void __launch_bounds__(256) kboxz_center(const float* __restrict__ in,
                                                    const float* __restrict__ orig,
                                                    float* __restrict__ cOut) {
  const int t = blockIdx.x * 8 + (threadIdx.x >> 5);
  int f0; const unsigned base = tile_decode(t, 160u, f0);
  const v8f acc = wave_box16_g(in, base, f0, HWu);
  const int n_ = threadIdx.x & 15, h = (threadIdx.x & 31) >> 4;
#pragma unroll
  for (int r = 0; r < 8; ++r) {
    const unsigned idx = base + (unsigned)(f0 + r + 8 * h) * HWu + (unsigned)n_;
    cOut[idx] = orig[idx] - acc[r] * SZ_INV;
  }
}

// ---- Z pass (both fields) + sigma + normalize + product: p = n_F * n_M
__global__ void __launch_bounds__(256) kboxz_norm(const float* __restrict__ tF,
                                                  const float* __restrict__ tM,
                                                  const float* __restrict__ cF,
                                                  const float* __restrict__ cM,
                                                  float* __restrict__ p) {
  const int t = blockIdx.x * 8 + (threadIdx.x >> 5);
  int f0; const unsigned base = tile_decode(t, 160u, f0);
  const v8f aF = wave_box16_g(tF, base, f0, HWu);
  const v8f aM = wave_box16_g(tM, base, f0, HWu);
  const int n_ = threadIdx.x & 15, h = (threadIdx.x & 31) >> 4;
#pragma unroll
  for (int r = 0; r < 8; ++r) {
    const unsigned idx = base + (unsigned)(f0 + r + 8 * h) * HWu + (unsigned)n_;
    const float sF = sqrtf(aF[r] * SZ_INV) + EPSV;
    const float sM = sqrtf(aM[r] * SZ_INV) + EPSV;
    p[idx] = (cF[idx] / sF) * (cM[idx] / sM);
  }
}

// ---- Z pass + loss accumulation: partial[block] = sum(cross^2 * mask)
__global__ void __launch_bounds__(256) kboxz_loss(const float* __restrict__ in,
                                                  const float* __restrict__ mask,
                                                  float* __restrict__ partials) {
  const int t = blockIdx.x * 8 + (threadIdx.x >> 5);
  int f0; const unsigned base = tile_decode(t, 160u, f0);
  const v8f acc = wave_box16_g(in, base, f0, HWu);
  const int n_ = threadIdx.x & 15, h = (threadIdx.x & 31) >> 4;
  float s = 0.0f;
#pragma unroll
  for (int r = 0; r < 8; ++r) {
    const unsigned idx = base + (unsigned)(f0 + r + 8 * h) * HWu + (unsigned)n_;
    const float cr = acc[r];
    s += cr * cr * mask[idx];
  }
  __shared__ float red[256];
  red[threadIdx.x] = s; __syncthreads();
  for (int off = 128; off > 0; off >>= 1) {
    if (threadIdx.x < (unsigned)off) red[threadIdx.x] += red[threadIdx.x + off];
    __syncthreads();
  }
  if (threadIdx.x == 0) partials[blockIdx.x] = red[0];
}

// ---- deterministic final reduction (single block, fixed order)
__global__ void __launch_bounds__(256) kfinal(const float* __restrict__ partials,
                                              int n, float* __restrict__ out) {
  float s = 0.0f;
  for (int i = threadIdx.x; i < n; i += 256) s += partials[i];
  __shared__ float red[256];
  red[threadIdx.x] = s; __syncthreads();
  for (int off = 128; off > 0; off >>= 1) {
    if (threadIdx.x < (unsigned)off) red[threadIdx.x] += red[threadIdx.x + off];
    __syncthreads();
  }
  if (threadIdx.x == 0) out[0] = -red[0];
}

extern "C" void kernel_launch(void* const* d_in, const int* in_sizes, int n_in,
                              void* d_out, int out_size, void* d_ws, size_t ws_size,
                              hipStream_t stream) {
  (void)in_sizes; (void)n_in; (void)out_size; (void)ws_size;
  const float* F    = (const float*)d_in[0];
  const float* M    = (const float*)d_in[1];
  const float* mask = (const float*)d_in[2];
  float* out = (float*)d_out;

  float* b0 = (float*)d_ws;             // 6 full volumes + partials in workspace
  float* b1 = b0 + TOTALu;
  float* b2 = b1 + TOTALu;
  float* b3 = b2 + TOTALu;
  float* b4 = b3 + TOTALu;
  float* b5 = b4 + TOTALu;
  float* partials = b5 + TOTALu;        // WBLOCKS floats

  // chain 1: local means -> centered fields c = x - u
  kbox_x    <<<X4BLOCKS, 256, 0, stream>>>(F, b0);
  kbox_x    <<<X4BLOCKS, 256, 0, stream>>>(M, b1);
  kbox_wmma <<<WBLOCKS, 256, 0, stream>>>(b0, b2, 160u, HWu);        // Y
  kbox_wmma <<<WBLOCKS, 256, 0, stream>>>(b1, b3, 160u, HWu);        // Y
  kboxz_center<<<WBLOCKS, 256, 0, stream>>>(b2, F, b0);              // Z -> c_F
  kboxz_center<<<WBLOCKS, 256, 0, stream>>>(b3, M, b1);              // Z -> c_M

  // chain 2: local sigmas -> p = n_F * n_M
  kbox_x_sq <<<X4BLOCKS, 256, 0, stream>>>(b0, b2);                  // X of c_F^2
  kbox_x_sq <<<X4BLOCKS, 256, 0, stream>>>(b1, b3);                  // X of c_M^2
  kbox_wmma <<<WBLOCKS, 256, 0, stream>>>(b2, b4, 160u, HWu);        // Y
  kbox_wmma <<<WBLOCKS, 256, 0, stream>>>(b3, b5, 160u, HWu);        // Y
  kboxz_norm<<<WBLOCKS, 256, 0, stream>>>(b4, b5, b0, b1, b2);       // Z -> p

  // chain 3: cross = box(p); loss = -sum(cross^2 * mask)
  kbox_x    <<<X4BLOCKS, 256, 0, stream>>>(b2, b3);                  // X
  kbox_wmma <<<WBLOCKS, 256, 0, stream>>>(b3, b4, 160u, HWu);        // Y
  kboxz_loss<<<WBLOCKS, 256, 0, stream>>>(b4, mask, partials);       // Z + partials
  kfinal    <<<1,       256, 0, stream>>>(partials, WBLOCKS, out);
}